// ParticleSystem_74174085202609
// MI455X (gfx1250) — compile-verified
//
#include <hip/hip_runtime.h>

#define NPART     8192
#define MAX_PAIRS 262144
#define NTILES    (NPART / 16)   // 512 tiles of 16 rows/cols

typedef float v2f __attribute__((ext_vector_type(2)));
typedef float v8f __attribute__((ext_vector_type(8)));

// ---------------------------------------------------------------------------
// K0: zero the output buffer, zero the per-row counters, and pack
// pos4[t] = (x, y, z, |p|^2) for aligned single-b128 loads in the sweeps.
// ---------------------------------------------------------------------------
__global__ void k_init(const float* __restrict__ pos, float* __restrict__ out,
                       int out_n, int* __restrict__ counts,
                       float4* __restrict__ pos4) {
    int t = blockIdx.x * blockDim.x + threadIdx.x;
    if (t < out_n) out[t] = 0.0f;
    if (t < NPART) {
        counts[t] = 0;
        float x = pos[3 * t + 0];
        float y = pos[3 * t + 1];
        float z = pos[3 * t + 2];
        pos4[t] = make_float4(x, y, z, x * x + y * y + z * z);
    }
}

// ---------------------------------------------------------------------------
// K1: count neighbours per row using V_WMMA_F32_16X16X4_F32 Gram tiles.
// One block per 16-row i-block; 8 waves stride over the 512 j-tiles.
// Lane t (<16) accumulates the count for row i0+t; flushed with atomicAdd.
// ---------------------------------------------------------------------------
__global__ void __launch_bounds__(256) k_count(const float4* __restrict__ pos4,
                                               const float* __restrict__ hp,
                                               int* __restrict__ counts) {
    const float h  = hp[0];
    const float h2 = h * h;
    const int i0   = blockIdx.x * 16;
    const unsigned l = threadIdx.x & 31u;
    const int wave = threadIdx.x >> 5;      // 0..7
    const int half = (l >= 16) ? 1 : 0;     // wave32 half selects K-pairs / M+8
    const int li   = (int)(l & 15u);

    // A matrix (16x4, K = x,y,z,0): lanes 0-15 hold K0/K1, lanes 16-31 K2/K3
    v2f a;
    {
        float4 pi = pos4[i0 + li];
        a.x = half ? pi.z : pi.x;
        a.y = half ? 0.0f : pi.y;
    }
    // |p_i|^2 for the 8 rows this lane's C registers correspond to
    float qi[8];
#pragma unroll
    for (int r = 0; r < 8; ++r) qi[r] = pos4[i0 + r + half * 8].w;

    int acc = 0;  // lane t<16: running count for row i0+t
    for (int jt = wave; jt < NTILES; jt += 8) {
        const int j = jt * 16 + li;
        const float4 pj = pos4[j];           // one aligned global_load_b128
        v2f b;
        b.x = half ? pj.z : pj.x;
        b.y = half ? 0.0f : pj.y;
        const float qj = pj.w;

        v8f c = {};
        c = __builtin_amdgcn_wmma_f32_16x16x4_f32(false, a, false, b,
                                                  (short)0, c, false, false);

        bool cond[8];
        bool any = false;
#pragma unroll
        for (int r = 0; r < 8; ++r) {
            float d2 = qi[r] + qj - 2.0f * c[r];
            int gi = i0 + r + half * 8;
            cond[r] = (d2 < h2) && (d2 > 0.0f) && (gi != j);
            any |= cond[r];
        }
        if (__builtin_amdgcn_ballot_w32(any) == 0u) continue;  // ~88% of tiles

#pragma unroll
        for (int r = 0; r < 8; ++r) {
            unsigned bl = __builtin_amdgcn_ballot_w32(cond[r]);
            if (bl == 0u) continue;
            int plo = __popc(bl & 0xFFFFu);  // row i0+r    (lanes 0-15)
            int phi = __popc(bl >> 16);      // row i0+r+8  (lanes 16-31)
            if ((int)l == r)     acc += plo;
            if ((int)l == r + 8) acc += phi;
        }
    }
    if (l < 16u && acc) atomicAdd(&counts[i0 + li], acc);
}

// ---------------------------------------------------------------------------
// K2: exclusive prefix sum over the 8192 row counts (single 1024-thread block)
// ---------------------------------------------------------------------------
__global__ void __launch_bounds__(1024) k_scan(const int* __restrict__ counts,
                                               int* __restrict__ offs) {
    __shared__ int sh[1024];
    const int t = threadIdx.x;
    const int base = t * 8;
    int v[8];
    int s = 0;
#pragma unroll
    for (int k = 0; k < 8; ++k) { v[k] = counts[base + k]; s += v[k]; }
    sh[t] = s;
    __syncthreads();
    for (int off = 1; off < 1024; off <<= 1) {
        int add = (t >= off) ? sh[t - off] : 0;
        __syncthreads();
        sh[t] += add;
        __syncthreads();
    }
    int ex = sh[t] - s;  // exclusive base for this thread's 8 rows
#pragma unroll
    for (int k = 0; k < 8; ++k) { offs[base + k] = ex; ex += v[k]; }
}

// ---------------------------------------------------------------------------
// K3: emit pass. One wave owns a 16-row i-block, walks j-tiles in ascending
// order (same WMMA arithmetic as K1 -> identical classification), and writes
// hits with ballot-prefix ranks, reproducing jnp.where's flat-index order.
// ---------------------------------------------------------------------------
__global__ void __launch_bounds__(32) k_write(const float4* __restrict__ pos4,
                                              const float* __restrict__ hp,
                                              const int* __restrict__ offs,
                                              float* __restrict__ out) {
    const float h  = hp[0];
    const float h2 = h * h;
    const int i0   = blockIdx.x * 16;
    const unsigned l = threadIdx.x & 31u;
    const int half = (l >= 16) ? 1 : 0;
    const int li   = (int)(l & 15u);

    v2f a;
    {
        float4 pi = pos4[i0 + li];
        a.x = half ? pi.z : pi.x;
        a.y = half ? 0.0f : pi.y;
    }
    float qi[8];
#pragma unroll
    for (int r = 0; r < 8; ++r) qi[r] = pos4[i0 + r + half * 8].w;

    // lane t<16 holds the running output slot for row i0+t
    int rowoff = (l < 16u) ? offs[i0 + li] : 0;

    float* outI = out;
    float* outJ = out + MAX_PAIRS;
    float* outR = out + 2 * MAX_PAIRS;

    for (int jt = 0; jt < NTILES; ++jt) {
        const int j = jt * 16 + li;
        const float4 pj = pos4[j];           // one aligned global_load_b128
        v2f b;
        b.x = half ? pj.z : pj.x;
        b.y = half ? 0.0f : pj.y;
        const float qj = pj.w;

        v8f c = {};
        c = __builtin_amdgcn_wmma_f32_16x16x4_f32(false, a, false, b,
                                                  (short)0, c, false, false);

        bool cond[8];
        bool any = false;
#pragma unroll
        for (int r = 0; r < 8; ++r) {
            float d2 = qi[r] + qj - 2.0f * c[r];
            int gi = i0 + r + half * 8;
            cond[r] = (d2 < h2) && (d2 > 0.0f) && (gi != j);
            any |= cond[r];
        }
        if (__builtin_amdgcn_ballot_w32(any) == 0u) continue;

#pragma unroll
        for (int r = 0; r < 8; ++r) {
            unsigned bl = __builtin_amdgcn_ballot_w32(cond[r]);
            if (bl == 0u) continue;
            unsigned lo = bl & 0xFFFFu;
            unsigned hi = bl >> 16;
            const int myrow = r + half * 8;
            const int robase = __shfl(rowoff, myrow, 32);  // row's current slot
            if (cond[r]) {
                unsigned sub = half ? hi : lo;
                int rank = __popc(sub & ((1u << li) - 1u));
                int s = robase + rank;
                if (s < MAX_PAIRS) {
                    int gi = i0 + myrow;
                    float4 pi = pos4[gi];
                    outI[s] = (float)gi;
                    outJ[s] = (float)j;
                    outR[3 * s + 0] = pj.x - pi.x;   // r_ij = pos[j] - pos[i]
                    outR[3 * s + 1] = pj.y - pi.y;
                    outR[3 * s + 2] = pj.z - pi.z;
                }
            }
            int plo = __popc(lo), phi = __popc(hi);
            if ((int)l == r)     rowoff += plo;
            if ((int)l == r + 8) rowoff += phi;
        }
    }
}

// ---------------------------------------------------------------------------
extern "C" void kernel_launch(void* const* d_in, const int* in_sizes, int n_in,
                              void* d_out, int out_size, void* d_ws, size_t ws_size,
                              hipStream_t stream) {
    const float* pos = (const float*)d_in[0];
    const float* h   = (const float*)d_in[1];
    float* out = (float*)d_out;

    // workspace layout: counts[8192] i32 | offsets[8192] i32 | pos4[8192] float4
    int*    counts = (int*)d_ws;
    int*    offs   = counts + NPART;
    float4* pos4   = (float4*)(offs + NPART);

    const int initN = out_size;  // 5 * MAX_PAIRS floats
    const int initB = (initN + 255) / 256;
    hipLaunchKernelGGL(k_init, dim3(initB), dim3(256), 0, stream,
                       pos, out, initN, counts, pos4);
    hipLaunchKernelGGL(k_count, dim3(NTILES), dim3(256), 0, stream,
                       pos4, h, counts);
    hipLaunchKernelGGL(k_scan, dim3(1), dim3(1024), 0, stream, counts, offs);
    hipLaunchKernelGGL(k_write, dim3(NTILES), dim3(32), 0, stream,
                       pos4, h, offs, out);
}